// EdgeDecoder_26706106646646
// MI455X (gfx1250) — compile-verified
//
#include <hip/hip_runtime.h>
#include <hip/hip_bf16.h>

typedef __attribute__((ext_vector_type(2))) float v2f;
typedef __attribute__((ext_vector_type(8))) float v8f;

// ---------------------------------------------------------------------------
// ew[e] = weight[e] * sim[e]
// ---------------------------------------------------------------------------
__global__ void ew_kernel(const float* __restrict__ w, const float* __restrict__ s,
                          float* __restrict__ ew, int E) {
    int i = blockIdx.x * blockDim.x + threadIdx.x;
    if (i < E) ew[i] = w[i] * s[i];
}

// ---------------------------------------------------------------------------
// zero a float buffer
// ---------------------------------------------------------------------------
__global__ void zero_kernel(float* __restrict__ p, long long n) {
    long long i = (long long)blockIdx.x * blockDim.x + threadIdx.x;
    if (i < n) p[i] = 0.0f;
}

// ---------------------------------------------------------------------------
// Edge scatter: agg[dst] += x[src] * ew  (float4 per thread, 4 f32 atomics)
// Threads: E * (Cin/4); adjacent threads cover channels of the same edge
// so the x[src] gathers are coalesced 128B bursts.
// ---------------------------------------------------------------------------
__global__ void scatter_kernel(const float* __restrict__ x,
                               const int* __restrict__ src,
                               const int* __restrict__ dst,
                               const float* __restrict__ ew,
                               float* __restrict__ agg,
                               int cin_div4, long long total) {
    long long tid = (long long)blockIdx.x * blockDim.x + threadIdx.x;
    if (tid >= total) return;
    int e  = (int)(tid / cin_div4);
    int c4 = (int)(tid % cin_div4);
    int s = src[e];
    int d = dst[e];
    float w = ew[e];
    const float4* xp = (const float4*)x + (long long)s * cin_div4 + c4;
    float4 v = *xp;
    float* ap = agg + ((long long)d * cin_div4 + c4) * 4;
    atomicAdd(ap + 0, v.x * w);
    atomicAdd(ap + 1, v.y * w);
    atomicAdd(ap + 2, v.z * w);
    atomicAdd(ap + 3, v.w * w);
}

// ---------------------------------------------------------------------------
// Fused dense layer: Y = act( Agg @ Wc + X @ Wl + bc + bl )
// One wave = one 16x16 output tile, accumulated with V_WMMA_F32_16X16X4_F32.
//
// A 16x4 f32 layout  (ISA 7.12.2): lane = 16*h + m  -> row M=m, v2f = {K=2h, K=2h+1}
// B  4x16 f32 layout (symmetric):  lane = 16*h + n  -> col N=n, v2f = {K=2h, K=2h+1}
// C/D 16x16 f32:                   acc[r] is row (h*8 + r), col n.
// Column guard for Cout=75 uses clamped addresses * mask (EXEC stays all-1s).
// ---------------------------------------------------------------------------
__global__ void gemm_fused_kernel(const float* __restrict__ Agg,
                                  const float* __restrict__ X,
                                  const float* __restrict__ Wc,
                                  const float* __restrict__ Wl,
                                  const float* __restrict__ bc,
                                  const float* __restrict__ bl,
                                  float* __restrict__ Y,
                                  int Cin, int Cout,
                                  int coutTiles, int totalTiles,
                                  int relu) {
    int wave = blockIdx.x * (blockDim.x >> 5) + (threadIdx.x >> 5);
    if (wave >= totalTiles) return;   // wave-uniform: EXEC all-1s inside

    int tm = wave / coutTiles;
    int tn = wave - tm * coutTiles;

    int lane  = threadIdx.x & 31;
    int half  = lane >> 4;        // K-half selector (0/1)
    int idx16 = lane & 15;

    int arow = tm * 16 + idx16;   // global row for A operand (N % 16 == 0, always valid)
    int n    = tn * 16 + idx16;   // global output column for B/C/D
    int nc   = n < Cout ? n : (Cout - 1);
    float bmask = n < Cout ? 1.0f : 0.0f;

    const float* __restrict__ a0 = Agg + (size_t)arow * Cin;
    const float* __restrict__ a1 = X   + (size_t)arow * Cin;

    v8f acc = {};

    // Agg @ Wc
    #pragma unroll 4
    for (int k0 = 0; k0 < Cin; k0 += 4) {
        int kk = k0 + half * 2;
        v2f a; a.x = a0[kk]; a.y = a0[kk + 1];
        v2f b;
        b.x = Wc[(size_t)kk * Cout + nc] * bmask;
        b.y = Wc[(size_t)(kk + 1) * Cout + nc] * bmask;
        acc = __builtin_amdgcn_wmma_f32_16x16x4_f32(false, a, false, b,
                                                    (short)0, acc, false, false);
    }
    // X @ Wl
    #pragma unroll 4
    for (int k0 = 0; k0 < Cin; k0 += 4) {
        int kk = k0 + half * 2;
        v2f a; a.x = a1[kk]; a.y = a1[kk + 1];
        v2f b;
        b.x = Wl[(size_t)kk * Cout + nc] * bmask;
        b.y = Wl[(size_t)(kk + 1) * Cout + nc] * bmask;
        acc = __builtin_amdgcn_wmma_f32_16x16x4_f32(false, a, false, b,
                                                    (short)0, acc, false, false);
    }

    float bias = bc[nc] + bl[nc];
    if (n < Cout) {
        int mbase = tm * 16 + half * 8;
        #pragma unroll
        for (int r = 0; r < 8; ++r) {
            float v = acc[r] + bias;
            if (relu) v = fmaxf(v, 0.0f);
            Y[(size_t)(mbase + r) * Cout + n] = v;
        }
    }
}

// ---------------------------------------------------------------------------
// Driver
// ---------------------------------------------------------------------------
static inline int cdiv(long long a, long long b) { return (int)((a + b - 1) / b); }

static void run_layer(const float* x, const int* src, const int* dst, const float* ew,
                      float* agg, const float* Wc, const float* bc,
                      const float* Wl, const float* bl, float* y,
                      int N, int E, int Cin, int Cout, int relu, hipStream_t stream) {
    // zero agg
    long long aggN = (long long)N * Cin;
    zero_kernel<<<cdiv(aggN, 256), 256, 0, stream>>>(agg, aggN);
    // scatter
    long long total = (long long)E * (Cin / 4);
    scatter_kernel<<<cdiv(total, 256), 256, 0, stream>>>(x, src, dst, ew, agg,
                                                         Cin / 4, total);
    // fused dense transform (WMMA)
    int rowTiles  = N / 16;                 // N=100000 divisible by 16
    int coutTiles = (Cout + 15) / 16;
    int totalTiles = rowTiles * coutTiles;
    int blocks = (totalTiles + 7) / 8;      // 8 waves / 256-thread block
    gemm_fused_kernel<<<blocks, 256, 0, stream>>>(agg, x, Wc, Wl, bc, bl, y,
                                                  Cin, Cout, coutTiles, totalTiles, relu);
}

extern "C" void kernel_launch(void* const* d_in, const int* in_sizes, int n_in,
                              void* d_out, int out_size, void* d_ws, size_t ws_size,
                              hipStream_t stream) {
    const float* z      = (const float*)d_in[0];
    const int*   eidx   = (const int*)d_in[1];
    const float* weight = (const float*)d_in[2];
    const float* sim    = (const float*)d_in[3];
    const float* Wc0 = (const float*)d_in[4];  const float* bc0 = (const float*)d_in[5];
    const float* Wl0 = (const float*)d_in[6];  const float* bl0 = (const float*)d_in[7];
    const float* Wc1 = (const float*)d_in[8];  const float* bc1 = (const float*)d_in[9];
    const float* Wl1 = (const float*)d_in[10]; const float* bl1 = (const float*)d_in[11];
    const float* Wc2 = (const float*)d_in[12]; const float* bc2 = (const float*)d_in[13];
    const float* Wl2 = (const float*)d_in[14]; const float* bl2 = (const float*)d_in[15];

    const int E = in_sizes[2];
    const int N = in_sizes[0] / 64;
    const int* src = eidx;
    const int* dst = eidx + E;

    // workspace carve-up (f32 elements), 256B-aligned chunks
    float* ws   = (float*)d_ws;
    size_t off  = 0;
    float* ew   = ws + off; off += ((size_t)E + 63) & ~(size_t)63;
    float* agg  = ws + off; off += (size_t)N * 128;
    float* buf1 = ws + off; off += (size_t)N * 128;
    float* buf2 = ws + off;

    // per-edge coefficient
    ew_kernel<<<cdiv(E, 256), 256, 0, stream>>>(weight, sim, ew, E);

    // layer 0: 64 -> 128, relu
    run_layer(z,    src, dst, ew, agg, Wc0, bc0, Wl0, bl0, buf1, N, E, 64, 128, 1, stream);
    // layer 1: 128 -> 128, relu
    run_layer(buf1, src, dst, ew, agg, Wc1, bc1, Wl1, bl1, buf2, N, E, 128, 128, 1, stream);
    // layer 2: 128 -> 75, no relu, straight to output
    run_layer(buf2, src, dst, ew, agg, Wc2, bc2, Wl2, bl2, (float*)d_out,
              N, E, 128, 75, 0, stream);
}